// RoiPooling_51556787421274
// MI455X (gfx1250) — compile-verified
//
#include <hip/hip_runtime.h>
#include <stdint.h>

#define POOL 7
#define FH   14
#define FW   14
#define CCH  1024
#define C4   256          // CCH / 4 float4 chunks
#define SCALE 0.0625f     // FH / IMG = 14/224

typedef float v4f __attribute__((ext_vector_type(4)));

__device__ __forceinline__ int iclampi(int v, int lo, int hi) {
  v = v < lo ? lo : v;
  return v > hi ? hi : v;
}

__global__ __launch_bounds__(256, 4) void roi_bilinear_pool(
    const float* __restrict__ img,   // (B, 14, 14, 1024) f32
    const float* __restrict__ rois,  // (B, 4) f32: ymin,xmin,ymax,xmax in pixels
    float* __restrict__ out)         // (B, 7, 7, 1024) f32
{
  __shared__ v4f buf[2][4][C4];      // 32 KB: double-buffered 4 gathered vectors

  const int py = blockIdx.x;         // 0..6  output row
  const int b  = blockIdx.y;         // batch
  const int t  = threadIdx.x;        // 0..255 channel-vector id

  // ---- ROI coordinate math (uniform across block, matches reference) ----
  const float r0 = rois[4*b+0];
  const float r1 = rois[4*b+1];
  const float r2 = rois[4*b+2];
  const float r3 = rois[4*b+3];

  const int ymin = iclampi((int)floorf(r0 * SCALE), 0, FH-1);
  const int xmin = iclampi((int)floorf(r1 * SCALE), 0, FW-1);
  const int ymax = iclampi((int)floorf(r2 * SCALE), ymin, FH-1);
  const int xmax = iclampi((int)floorf(r3 * SCALE), xmin, FW-1);

  const int   dy    = ymax - ymin;
  const float stepy = (float)(dy + 1) / (float)POOL;
  const float sy    = (float)py * stepy;
  const int   iy0   = (int)sy;
  const int   iy1   = (iy0 + 1 < dy) ? (iy0 + 1) : dy;   // min(i0+1, hi-lo)
  const float ly    = sy - (float)iy0;

  const int   dxr   = xmax - xmin;
  const float stepx = (float)(dxr + 1) / (float)POOL;
  int   xl[POOL], xh[POOL];
  float lx[POOL];
#pragma unroll
  for (int j = 0; j < POOL; ++j) {
    float sx = (float)j * stepx;
    int   i0 = (int)sx;
    int   i1 = (i0 + 1 < dxr) ? (i0 + 1) : dxr;
    xl[j] = xmin + i0;
    xh[j] = xmin + i1;
    lx[j] = sx - (float)i0;
  }

  // Row base pointers for this thread's channel chunk (contiguous along C).
  const float* row0 = img + ((size_t)(b * FH + (ymin + iy0)) * FW) * CCH + 4 * t;
  const float* row1 = img + ((size_t)(b * FH + (ymin + iy1)) * FW) * CCH + 4 * t;

  // ---- async gather pipeline: CDNA5 GLOBAL_LOAD_ASYNC_TO_LDS (ASYNCcnt) ----
  auto issue = [&](int px, int pb) {
    // WAR: my previous LDS reads of this buffer must be done before the
    // async engine overwrites it (ds pipe and async pipe are unordered).
    asm volatile("s_wait_dscnt 0x0" ::: "memory");
    unsigned l0 = (unsigned)(uintptr_t)(&buf[pb][0][t]);
    unsigned l1 = (unsigned)(uintptr_t)(&buf[pb][1][t]);
    unsigned l2 = (unsigned)(uintptr_t)(&buf[pb][2][t]);
    unsigned l3 = (unsigned)(uintptr_t)(&buf[pb][3][t]);
    unsigned long long g00 = (unsigned long long)(uintptr_t)(row0 + (size_t)xl[px] * CCH);
    unsigned long long g01 = (unsigned long long)(uintptr_t)(row0 + (size_t)xh[px] * CCH);
    unsigned long long g10 = (unsigned long long)(uintptr_t)(row1 + (size_t)xl[px] * CCH);
    unsigned long long g11 = (unsigned long long)(uintptr_t)(row1 + (size_t)xh[px] * CCH);
    asm volatile("global_load_async_to_lds_b128 %0, %1, off" :: "v"(l0), "v"(g00) : "memory");
    asm volatile("global_load_async_to_lds_b128 %0, %1, off" :: "v"(l1), "v"(g01) : "memory");
    asm volatile("global_load_async_to_lds_b128 %0, %1, off" :: "v"(l2), "v"(g10) : "memory");
    asm volatile("global_load_async_to_lds_b128 %0, %1, off" :: "v"(l3), "v"(g11) : "memory");
  };

  auto consume = [&](int px, int pb) {
    v4f v00 = buf[pb][0][t];   // (y0, x0)
    v4f v01 = buf[pb][1][t];   // (y0, x1)
    v4f v10 = buf[pb][2][t];   // (y1, x0)
    v4f v11 = buf[pb][3][t];   // (y1, x1)
    float fx = lx[px];
    v4f top = v00 + (v01 - v00) * fx;
    v4f bot = v10 + (v11 - v10) * fx;
    v4f res = top + (bot - top) * ly;
    float* dst = out + (((size_t)(b * POOL + py) * POOL + px) * CCH) + 4 * t;
    __builtin_nontemporal_store(res, (v4f*)dst);   // write-once stream: keep L2 for fmap
  };

  issue(0, 0);
#pragma unroll
  for (int px = 0; px < POOL; ++px) {
    if (px + 1 < POOL) {
      issue(px + 1, (px + 1) & 1);
      // 8 async loads outstanding; <=4 left means group px landed in LDS
      // (async loads complete in order).
      asm volatile("s_wait_asynccnt 0x4" ::: "memory");
    } else {
      asm volatile("s_wait_asynccnt 0x0" ::: "memory");
    }
    consume(px, px & 1);
  }
}

extern "C" void kernel_launch(void* const* d_in, const int* in_sizes, int n_in,
                              void* d_out, int out_size, void* d_ws, size_t ws_size,
                              hipStream_t stream) {
  const float* img  = (const float*)d_in[0];   // (B,14,14,1024) f32
  const float* rois = (const float*)d_in[1];   // (B,4) f32
  float* out = (float*)d_out;                  // (B,7,7,1024) f32
  const int B = in_sizes[1] / 4;
  dim3 grid(POOL, (unsigned)B);
  roi_bilinear_pool<<<grid, 256, 0, stream>>>(img, rois, out);
}